// CPLayer_18726057410945
// MI455X (gfx1250) — compile-verified
//
#include <hip/hip_runtime.h>

// CP gate layer: out = U @ psi (complex, split re/im), U = dense expansion of a
// diagonal gate. Exploit block-diagonal structure: only the 256 diagonal 16x16
// blocks of U contribute; compute them with V_WMMA_F32_16X16X4_F32 (bit-exact
// f32). Traffic ~4.5 MB -> ~0.2us at 23.3 TB/s vs ~5.8us for the dense GEMM.

typedef __attribute__((ext_vector_type(2))) float v2f;
typedef __attribute__((ext_vector_type(4))) float v4f;
typedef __attribute__((ext_vector_type(8))) float v8f;

#define DIMN   4096
#define BATCHN 64

__global__ __launch_bounds__(256) void cp_blockdiag_wmma_f32(
    const float* __restrict__ U_re, const float* __restrict__ U_im,
    const float* __restrict__ psi_re, const float* __restrict__ psi_im,
    float* __restrict__ out)
{
    const int lane = threadIdx.x & 31;
    const int wid  = threadIdx.x >> 5;               // 8 waves per block
    const int ktile = blockIdx.x * 2 + (wid >> 2);   // 256 k-tiles total
    const int btile = wid & 3;                       // 4 batch-tiles total
    const int k0 = ktile << 4;
    const int b0 = btile << 4;

    const int m  = lane & 15;   // A row / B col / D col within tile
    const int hi = lane >> 4;   // half-wave select (K pair / M+8)

    const size_t Dz = (size_t)DIMN;

    // ---- A fragments: 16x16 diagonal block of U, as four 16x4 K-chunks.
    // f32 A layout: lanes 0-15 hold K={0,1} in vgpr{0,1}; lanes 16-31 hold K={2,3}.
    const float* Ar = U_re + (size_t)(k0 + m) * Dz + (size_t)(k0 + hi * 2);
    const float* Ai = U_im + (size_t)(k0 + m) * Dz + (size_t)(k0 + hi * 2);
    v2f a_re[4], a_im[4];
#pragma unroll
    for (int c = 0; c < 4; ++c) {
        a_re[c] = *(const v2f*)(Ar + 4 * c);   // 8B-aligned global_load_b64
        a_im[c] = *(const v2f*)(Ai + 4 * c);
    }

    // ---- B fragments: P[j,b] = psi[b*D + j]; j along K, batch along N.
    // psi is batch-major, so j is contiguous per lane -> direct v2f loads.
    const float* Pr = psi_re + (size_t)(b0 + m) * Dz + (size_t)(k0 + hi * 2);
    const float* Pi = psi_im + (size_t)(b0 + m) * Dz + (size_t)(k0 + hi * 2);
    v2f b_re[4], b_im[4], b_imn[4];
#pragma unroll
    for (int c = 0; c < 4; ++c) {
        b_re[c] = *(const v2f*)(Pr + 4 * c);
        b_im[c] = *(const v2f*)(Pi + 4 * c);
        b_imn[c].x = -b_im[c].x;               // f32 WMMA has no B-negate (NEG=CNeg only)
        b_imn[c].y = -b_im[c].y;
    }

    // ---- Complex block-GEMM:
    //   re = Ur*Pr - Ui*Pi ; im = Ur*Pi + Ui*Pr
    v8f acc_re = {};
    v8f acc_im = {};
#pragma unroll
    for (int c = 0; c < 4; ++c) {
        acc_re = __builtin_amdgcn_wmma_f32_16x16x4_f32(false, a_re[c], false, b_re[c],  (short)0, acc_re, false, false);
        acc_re = __builtin_amdgcn_wmma_f32_16x16x4_f32(false, a_im[c], false, b_imn[c], (short)0, acc_re, false, false);
        acc_im = __builtin_amdgcn_wmma_f32_16x16x4_f32(false, a_re[c], false, b_im[c],  (short)0, acc_im, false, false);
        acc_im = __builtin_amdgcn_wmma_f32_16x16x4_f32(false, a_im[c], false, b_re[c],  (short)0, acc_im, false, false);
    }

    // ---- D layout: vgpr v, lane<16 -> (M=v, N=lane); lane>=16 -> (M=v+8, N=lane-16).
    // M is k (contiguous per lane), N is batch. out[plane][b][k], plane stride B*D.
    const int bcol = b0 + m;
    const int krow = k0 + hi * 8;
    float* o_re = out + (size_t)bcol * Dz + krow;
    float* o_im = out + (size_t)BATCHN * Dz + (size_t)bcol * Dz + krow;
    const v4f* pre = (const v4f*)&acc_re;
    const v4f* pim = (const v4f*)&acc_im;
    *(v4f*)(o_re)     = pre[0];   // 32B-aligned b128 stores
    *(v4f*)(o_re + 4) = pre[1];
    *(v4f*)(o_im)     = pim[0];
    *(v4f*)(o_im + 4) = pim[1];
}

extern "C" void kernel_launch(void* const* d_in, const int* in_sizes, int n_in,
                              void* d_out, int out_size, void* d_ws, size_t ws_size,
                              hipStream_t stream) {
    (void)in_sizes; (void)n_in; (void)out_size; (void)d_ws; (void)ws_size;
    const float* U_re   = (const float*)d_in[0];
    const float* U_im   = (const float*)d_in[1];
    const float* psi_re = (const float*)d_in[2];
    const float* psi_im = (const float*)d_in[3];
    float* out = (float*)d_out;

    // 256 k-tiles x 4 b-tiles = 1024 waves, packed 8 waves (2 k-tiles x 4 b-tiles)
    // per 256-thread block -> 128 blocks.
    cp_blockdiag_wmma_f32<<<dim3(DIMN / 16 / 2), dim3(256), 0, stream>>>(
        U_re, U_im, psi_re, psi_im, out);
}